// RelGraphEmbedLayerHomo_25890062860616
// MI455X (gfx1250) — compile-verified
//
#include <hip/hip_runtime.h>

typedef float v2f __attribute__((ext_vector_type(2)));
typedef float v8f __attribute__((ext_vector_type(8)));

#define DOUT 128
#define DIN  256
#define TILE_M 64
#define LDS_STRIDE 260   // 256 + 4 pad: bank stride 4 -> conflict-free b64 A loads

// ---------------------------------------------------------------------------
// Kernel 1: classify nodes; copy embedding rows for featureless nodes,
// compact featured node ids into a global list (wave-aggregated atomics).
// ---------------------------------------------------------------------------
__global__ __launch_bounds__(256) void split_copy_kernel(
    const float* __restrict__ emb_table,
    const int*   __restrict__ node_ids,
    const int*   __restrict__ node_tids,
    float*       __restrict__ out,
    unsigned*    __restrict__ counter,
    int*         __restrict__ feat_list,
    int N)
{
    __shared__ int s_emb[256];
    __shared__ int s_cnt;
    if (threadIdx.x == 0) s_cnt = 0;
    __syncthreads();

    int g = blockIdx.x * 256 + threadIdx.x;
    unsigned lane = threadIdx.x & 31u;
    bool in = (g < N);
    int t = in ? node_tids[g] : 1;
    bool featured = in && (t == 0);
    bool plain    = in && (t != 0);
    unsigned lt = (1u << lane) - 1u;

    // featured -> global compacted list (one atomic per wave)
    unsigned fm = __builtin_amdgcn_ballot_w32(featured);
    if (fm) {
        unsigned b0 = 0;
        if (lane == 0) b0 = atomicAdd(counter, (unsigned)__popc(fm));
        unsigned base = (unsigned)__builtin_amdgcn_readlane((int)b0, 0);
        if (featured) feat_list[base + __popc(fm & lt)] = g;
    }
    // featureless -> LDS queue for this block
    unsigned em = __builtin_amdgcn_ballot_w32(plain);
    if (em) {
        int b0 = 0;
        if (lane == 0) b0 = atomicAdd(&s_cnt, __popc(em));
        int base = __builtin_amdgcn_readlane(b0, 0);
        if (plain) s_emb[base + __popc(em & lt)] = g;
    }
    __syncthreads();

    // each wave copies one 512B embedding row per iteration (float4/lane)
    int w = threadIdx.x >> 5;
    int cnt = s_cnt;
    for (int n = w; n < cnt; n += 8) {
        int gn = s_emb[n];
        long nid = node_ids[gn];
        const float4* src = (const float4*)(emb_table + nid * DOUT);
        float4*       dst = (float4*)(out + (long)gn * DOUT);
        dst[lane] = src[lane];
    }
}

// ---------------------------------------------------------------------------
// Kernel 2: tiles of 64 featured rows. Gather feat0 rows to LDS, then
// 8 waves x (16x16 out tile each, 4 M-subtiles) via v_wmma_f32_16x16x4_f32.
// ---------------------------------------------------------------------------
__global__ __launch_bounds__(256) void feat_gemm_kernel(
    const float* __restrict__ feat0,
    const float* __restrict__ W0,
    const int*   __restrict__ type_ids,
    float*       __restrict__ out,
    const unsigned* __restrict__ counter,
    const int*   __restrict__ feat_list)
{
    __shared__ float As[TILE_M][LDS_STRIDE];
    __shared__ int sIdx[TILE_M];
    __shared__ int sFid[TILE_M];

    int count = (int)*counter;
    int tile  = blockIdx.x * TILE_M;
    int valid = count - tile;
    if (valid <= 0) return;           // block-uniform
    if (valid > TILE_M) valid = TILE_M;

    int tid = threadIdx.x;
    if (tid < TILE_M) {
        int g = 0, f = 0;
        if (tid < valid) { g = feat_list[tile + tid]; f = type_ids[g]; }
        sIdx[tid] = g;
        sFid[tid] = f;
    }
    __syncthreads();

    // gather 64 rows x 256 f32 into LDS (zero-fill tail rows so WMMA is safe)
    int rsub = tid >> 6;          // 0..3
    int c4   = (tid & 63) << 2;   // 0..252
    for (int r0 = 0; r0 < TILE_M; r0 += 4) {
        int row = r0 + rsub;
        float4 v = make_float4(0.f, 0.f, 0.f, 0.f);
        if (row < valid)
            v = *(const float4*)(feat0 + (long)sFid[row] * DIN + c4);
        *(float4*)&As[row][c4] = v;
    }
    __syncthreads();

    int w    = tid >> 5;
    int lane = tid & 31;
    int hi2  = lane >> 4;         // 0: K pair {0,1}, 1: K pair {2,3}
    int coll = lane & 15;         // A row / C column
    int colg = (w << 4) + coll;   // global output column
    int koff = hi2 << 1;

    v8f c0 = {}, c1 = {}, c2 = {}, c3 = {};
    #pragma unroll 4
    for (int k = 0; k < DIN; k += 4) {
        int kk = k + koff;
        v2f b;
        b.x = W0[(long)kk * DOUT + colg];
        b.y = W0[(long)(kk + 1) * DOUT + colg];
        v2f a;
        a = *(const v2f*)&As[ 0 + coll][kk];
        c0 = __builtin_amdgcn_wmma_f32_16x16x4_f32(false, a, false, b, (short)0, c0, false, false);
        a = *(const v2f*)&As[16 + coll][kk];
        c1 = __builtin_amdgcn_wmma_f32_16x16x4_f32(false, a, false, b, (short)0, c1, false, false);
        a = *(const v2f*)&As[32 + coll][kk];
        c2 = __builtin_amdgcn_wmma_f32_16x16x4_f32(false, a, false, b, (short)0, c2, false, false);
        a = *(const v2f*)&As[48 + coll][kk];
        c3 = __builtin_amdgcn_wmma_f32_16x16x4_f32(false, a, false, b, (short)0, c3, false, false);
    }

    // C layout: VGPR i -> M = i + 8*hi2, N = coll
    #pragma unroll
    for (int i = 0; i < 8; ++i) {
        int rb = i + 8 * hi2;
        if (rb      < valid) out[(long)sIdx[rb     ] * DOUT + colg] = c0[i];
        if (16 + rb < valid) out[(long)sIdx[16 + rb] * DOUT + colg] = c1[i];
        if (32 + rb < valid) out[(long)sIdx[32 + rb] * DOUT + colg] = c2[i];
        if (48 + rb < valid) out[(long)sIdx[48 + rb] * DOUT + colg] = c3[i];
    }
}

// ---------------------------------------------------------------------------
extern "C" void kernel_launch(void* const* d_in, const int* in_sizes, int n_in,
                              void* d_out, int out_size, void* d_ws, size_t ws_size,
                              hipStream_t stream) {
    const float* feat0     = (const float*)d_in[0];
    const float* W0        = (const float*)d_in[1];
    const float* emb_table = (const float*)d_in[2];
    const int*   node_ids  = (const int*)d_in[3];
    const int*   node_tids = (const int*)d_in[4];
    const int*   type_ids  = (const int*)d_in[5];
    float* out = (float*)d_out;
    int N = in_sizes[3];

    unsigned* counter  = (unsigned*)d_ws;
    int*      feat_list = (int*)((char*)d_ws + 16);

    hipMemsetAsync(d_ws, 0, 16, stream);

    int blocksA = (N + 255) / 256;
    split_copy_kernel<<<blocksA, 256, 0, stream>>>(
        emb_table, node_ids, node_tids, out, counter, feat_list, N);

    int blocksB = (N + TILE_M - 1) / TILE_M;   // worst case: all nodes featured
    feat_gemm_kernel<<<blocksB, 256, 0, stream>>>(
        feat0, W0, type_ids, out, counter, feat_list);
}